// CanonicalColorLoss_2113123909883
// MI455X (gfx1250) — compile-verified
//
#include <hip/hip_runtime.h>
#include <hip/hip_bf16.h>

#define BIG 1e30f

typedef __attribute__((ext_vector_type(2))) float v2f;
typedef __attribute__((ext_vector_type(8))) float v8f;

// ---------------------------------------------------------------------------
// ws layout (contiguous):
//   sum_x : float[B*8]
//   sum_y : float[B*8]
//   cnt   : int  [B*8]
//   pm    : u8   [N]      packed part-mask bits per point
// ---------------------------------------------------------------------------

__global__ void zero_ws(float* p, int n) {
    int t = blockIdx.x * blockDim.x + threadIdx.x;
    if (t < n) p[t] = 0.0f;   // also zeroes the int cnt region (bit pattern 0)
}

// Pack 8 mask bits per point into one byte; accumulate per-(object,part) counts.
// Assumes M == 8 and P % blockDim.x == 0 (each block lies inside one object).
__global__ void pack_masks(const int* __restrict__ mask, unsigned char* __restrict__ pm,
                           int* __restrict__ cnt, int P_) {
    __shared__ int scnt[8];
    const int t = threadIdx.x;
    if (t < 8) scnt[t] = 0;
    __syncthreads();
    const int pg = blockIdx.x * blockDim.x + t;   // global point id
    const int b  = pg / P_;
    const int p  = pg - b * P_;
    unsigned int byte = 0;
#pragma unroll
    for (int m = 0; m < 8; ++m) {
        int bit = (mask[(b * 8 + m) * P_ + p] != 0);
        byte |= (unsigned)bit << m;
        if (bit) atomicAdd(&scnt[m], 1);
    }
    pm[pg] = (unsigned char)byte;
    __syncthreads();
    if (t < 8) atomicAdd(&cnt[b * 8 + t], scnt[t]);
}

// One pass of the directed chamfer term:
//   for rows i (X side): rowmin_m[i] = min_{j masked(m)} ||X_i - Y_j||
//   sums[b*8+m] += sum_{i masked(m)} rowmin_m[i]
// Launched twice with (X,Y) swapped to get both directions.
// grid = (P/128, nb), block = 256 (8 waves). Wave w owns rows i0..i0+15.
//
// All distances are >= 0, so float min == unsigned-int min of the bit
// patterns: accumulators are kept as u32 (v_min_u32, no NaN canonicalize).
__global__ void __launch_bounds__(256) chamfer_pass(
        const float* __restrict__ X, const float* __restrict__ Y,
        const int* __restrict__ pt_offset, const unsigned char* __restrict__ pm,
        float* __restrict__ sums, int P_) {
    const int b     = blockIdx.y;
    const int tid   = threadIdx.x;
    const int lane  = tid & 31;
    const int wave  = tid >> 5;
    const int lr    = lane & 15;     // column-within-tile / A-matrix M index
    const int hi    = lane >> 4;     // half select (K/M split per WMMA layout)
    const int base  = (b == 0) ? 0 : pt_offset[b - 1];  // data base (from pt_offset)
    const int mbase = b * P_;                            // mask base (by object)
    const int i0    = (blockIdx.x * 8 + wave) * 16;

    // A operand (16x4 f32): lane M = lr; VGPR0 = K0|K2, VGPR1 = K1|K3(pad=0)
    v2f a;
    {
        const float* xp = X + (size_t)(base + i0 + lr) * 3;
        float x0 = xp[0], x1 = xp[1], x2 = xp[2];
        a.x = hi ? x2 : x0;
        a.y = hi ? 0.0f : x1;
    }

    // Row norms + packed row-mask bytes for this lane's 8 D rows (r + 8*hi)
    float pn8[8];
    unsigned int rm8[8];
#pragma unroll
    for (int r = 0; r < 8; ++r) {
        int row = i0 + r + 8 * hi;
        const float* xp = X + (size_t)(base + row) * 3;
        pn8[r] = xp[0] * xp[0] + xp[1] * xp[1] + xp[2] * xp[2];
        rm8[r] = pm[mbase + row];
    }

    const unsigned int BIGU = __float_as_uint(BIG);
    unsigned int acc[8][8];
#pragma unroll
    for (int r = 0; r < 8; ++r)
#pragma unroll
        for (int m = 0; m < 8; ++m) acc[r][m] = BIGU;

    const int tiles = P_ >> 4;
#pragma unroll 2
    for (int jt = 0; jt < tiles; ++jt) {
        const int j = (jt << 4) + lr;
        const float* yp = Y + (size_t)(base + j) * 3;
        __builtin_prefetch(yp + 48, 0, 0);   // next 16-point tile -> global_prefetch_b8
        float y0 = yp[0], y1 = yp[1], y2 = yp[2];
        v2f bb;                               // B operand (4x16 f32), N = lr
        bb.x = hi ? y2 : y0;
        bb.y = hi ? 0.0f : y1;
        const float tn = y0 * y0 + y1 * y1 + y2 * y2;

        // Per-tile additive mask penalty: masked-out columns push dist to ~BIG.
        const unsigned int cmask = pm[mbase + j];
        float pen[8];
#pragma unroll
        for (int m = 0; m < 8; ++m)
            pen[m] = ((cmask >> m) & 1u) ? 0.0f : BIG;

        v8f c = {0.0f, 0.0f, 0.0f, 0.0f, 0.0f, 0.0f, 0.0f, 0.0f};
        // D[r] = dot(X_{i0+r+8*hi}, Y_{jt*16+lr}) via v_wmma_f32_16x16x4_f32
        v8f d = __builtin_amdgcn_wmma_f32_16x16x4_f32(
            false, a, false, bb, (short)0, c, false, false);

#pragma unroll
        for (int r = 0; r < 8; ++r) {
            float d2   = pn8[r] + tn - 2.0f * d[r];
            float dist = __builtin_amdgcn_sqrtf(fmaxf(d2, 0.0f));  // raw v_sqrt_f32
#pragma unroll
            for (int m = 0; m < 8; ++m) {
                unsigned int v = __float_as_uint(dist + pen[m]);
                acc[r][m] = (v < acc[r][m]) ? v : acc[r][m];       // v_min_u32
            }
        }
    }

    // Min-reduce across the 16 lanes of each half (each half holds one column set)
#pragma unroll
    for (int r = 0; r < 8; ++r)
#pragma unroll
        for (int m = 0; m < 8; ++m) {
            unsigned int v = acc[r][m];
            unsigned int w;
            w = (unsigned int)__shfl_xor((int)v, 1, 32); v = (w < v) ? w : v;
            w = (unsigned int)__shfl_xor((int)v, 2, 32); v = (w < v) ? w : v;
            w = (unsigned int)__shfl_xor((int)v, 4, 32); v = (w < v) ? w : v;
            w = (unsigned int)__shfl_xor((int)v, 8, 32); v = (w < v) ? w : v;
            acc[r][m] = v;
        }

    // Lanes 0 and 16 hold final row-mins for rows i0+r and i0+8+r respectively.
    if (lr == 0) {
#pragma unroll
        for (int m = 0; m < 8; ++m) {
            float s = 0.0f;
#pragma unroll
            for (int r = 0; r < 8; ++r)
                s += ((rm8[r] >> m) & 1u) ? __uint_as_float(acc[r][m]) : 0.0f;
            atomicAdd(&sums[b * 8 + m], s);
        }
    }
}

// Final reduction, mirrors the reference semantics exactly. 1 block, 128 threads.
__global__ void finalize(const float* __restrict__ sx, const float* __restrict__ sy,
                         const int* __restrict__ cnt, float* __restrict__ out, int nb) {
    __shared__ float ch_s[16 * 8];
    __shared__ float va_s[16 * 8];
    __shared__ float ol_s[16];
    __shared__ float ct_s[16];
    const int t = threadIdx.x;
    if (t < nb * 8) {
        int   n     = cnt[t];
        float denom = fmaxf((float)n, 1.0f);
        ch_s[t] = 0.5f * (sx[t] / denom + sy[t] / denom);
        va_s[t] = (n >= 2) ? 1.0f : 0.0f;
    }
    __syncthreads();
    if (t < nb) {
        float nv = 0.0f, tot = 0.0f;
        for (int m = 0; m < 8; ++m) {
            nv  += va_s[t * 8 + m];
            tot += ch_s[t * 8 + m] * va_s[t * 8 + m];
        }
        ol_s[t] = (nv > 0.0f) ? tot / fmaxf(nv, 1.0f) : 0.0f;
        ct_s[t] = (nv > 0.0f) ? 1.0f : 0.0f;
    }
    __syncthreads();
    if (t == 0) {
        float count = 0.0f, total = 0.0f;
        for (int b2 = 0; b2 < nb; ++b2) { count += ct_s[b2]; total += ol_s[b2] * ct_s[b2]; }
        out[0] = (count > 0.0f) ? total / fmaxf(count, 1.0f) : 0.0f;
    }
}

extern "C" void kernel_launch(void* const* d_in, const int* in_sizes, int n_in,
                              void* d_out, int out_size, void* d_ws, size_t ws_size,
                              hipStream_t stream) {
    const float* canon = (const float*)d_in[0];   // canoncolor_out [N,3] f32
    const float* gt    = (const float*)d_in[1];   // gt_color       [N,3] f32
    const int*   poff  = (const int*)d_in[2];     // pt_offset      [B]   i32
    const int*   mask  = (const int*)d_in[3];     // mask_pts       [B,8,P] (int)

    const int B_ = in_sizes[2];          // 16
    const int N_ = in_sizes[0] / 3;      // 49152
    const int P_ = N_ / B_;              // 3072
    const int nb = B_ - 1;               // 15 (reference skips last object)

    float* sum_x = (float*)d_ws;
    float* sum_y = sum_x + B_ * 8;
    int*   cnt   = (int*)(sum_y + B_ * 8);
    unsigned char* pm = (unsigned char*)(cnt + B_ * 8);

    const int zn = B_ * 8 * 3;  // sum_x, sum_y, cnt are contiguous
    zero_ws<<<(zn + 255) / 256, 256, 0, stream>>>(sum_x, zn);
    pack_masks<<<N_ / 256, 256, 0, stream>>>(mask, pm, cnt, P_);

    dim3 grid(P_ / 128, nb);
    chamfer_pass<<<grid, 256, 0, stream>>>(canon, gt, poff, pm, sum_x, P_);  // min over gt
    chamfer_pass<<<grid, 256, 0, stream>>>(gt, canon, poff, pm, sum_y, P_);  // min over pred

    finalize<<<1, 128, 0, stream>>>(sum_x, sum_y, cnt, (float*)d_out, nb);
}